// MprmMamba_1589137900243
// MI455X (gfx1250) — compile-verified
//
#include <hip/hip_runtime.h>
#include <math.h>

// ---- problem constants (from reference) ----
#define B_   2
#define NM_  4
#define E_   64
#define L_   4096     // 16^3 tokens per module
#define S_   32768    // 32^3 voxels
#define DI_  128
#define DS_  16
#define DC_  4
#define DTR_ 4
#define R_   32

typedef float v2f __attribute__((ext_vector_type(2)));
typedef float v8f __attribute__((ext_vector_type(8)));

// Device-pass-only check: __has_builtin for amdgcn builtins is 0 on the host pass.
#if defined(__HIP_DEVICE_COMPILE__) && !__has_builtin(__builtin_amdgcn_wmma_f32_16x16x4_f32)
#error "device: __builtin_amdgcn_wmma_f32_16x16x4_f32 not available on this toolchain"
#endif

__device__ __forceinline__ v8f wmma4(v2f a, v2f b, v8f c) {
#if defined(__HIP_DEVICE_COMPILE__)
  return __builtin_amdgcn_wmma_f32_16x16x4_f32(false, a, false, b, (short)0, c,
                                               false, false);
#else
  (void)a; (void)b;
  return c;  // host pass: parsed but never executed
#endif
}

__device__ __forceinline__ float siluf(float x) { return x / (1.f + __expf(-x)); }
__device__ __forceinline__ float softplusf(float x) { return (x > 20.f) ? x : log1pf(__expf(x)); }

// ---- fp32 WMMA helper: one wave computes a 16x16 tile, K in steps of 4 ----
// A: row-major MxK tile base (row tm*16 already applied), lda
// Bm: row-major KxN, column base tn*16 already applied, ldb
// ISA layout (7.12.2): lane l -> A[m=l&15][2*(l>>4)+v], B[2*(l>>4)+v][n=l&15],
//                      C/D: lane l, vgpr r -> [8*(l>>4)+r][l&15]
__device__ __forceinline__ v8f wmma_k_f32(const float* __restrict__ A, int lda,
                                          const float* __restrict__ Bm, int ldb,
                                          int K, v8f acc) {
  int lane = threadIdx.x & 31;
  int mr = lane & 15;
  int kh = lane >> 4;
  for (int k0 = 0; k0 < K; k0 += 4) {
    int ka = k0 + 2 * kh;
    v2f a, b;
    a.x = A[(size_t)mr * lda + ka];
    a.y = A[(size_t)mr * lda + ka + 1];
    b.x = Bm[(size_t)ka * ldb + mr];
    b.y = Bm[(size_t)(ka + 1) * ldb + mr];
    acc = wmma4(a, b, acc);
  }
  return acc;
}

__device__ __forceinline__ void wmma_store(float* __restrict__ C, int ldc, v8f acc) {
  int lane = threadIdx.x & 31;
  int n = lane & 15, mh = lane >> 4;
#pragma unroll
  for (int r = 0; r < 8; ++r)
    C[(size_t)(mh * 8 + r) * ldc + n] = acc[r];
}

// ---- K1: patchify + input projection: z[b,i,l,e] ----
__global__ void k_patch(const float* __restrict__ x0, const float* __restrict__ x1,
                        const float* __restrict__ x2, const float* __restrict__ x3,
                        const float* __restrict__ in_W, const float* __restrict__ in_b,
                        float* __restrict__ Z) {
  int l = blockIdx.x, i = blockIdx.y, b = blockIdx.z;
  int e = threadIdx.x;
  const float* xs[4] = {x0, x1, x2, x3};
  __shared__ float t[8];
  int gz = l >> 8, gy = (l >> 4) & 15, gx = l & 15;
  if (e < 8) {
    int dz = e >> 2, dy = (e >> 1) & 1, dx = e & 1;
    int zz = gz * 2 + dz, yy = gy * 2 + dy, xx = gx * 2 + dx;
    t[e] = xs[i][(size_t)b * S_ + zz * 1024 + yy * 32 + xx];
  }
  __syncthreads();
  float s = in_b[i * E_ + e];
#pragma unroll
  for (int j = 0; j < 8; ++j) s += t[j] * in_W[(i * 8 + j) * E_ + e];
  Z[(((size_t)(b * NM_ + i)) * L_ + l) * E_ + e] = s;
}

// ---- K2: descriptors d[b,i,e] = mean_l z ----
__global__ void k_desc(const float* __restrict__ Z, float* __restrict__ DESC) {
  int i = blockIdx.x, b = blockIdx.y;
  int bi = b * NM_ + i;
  int e = threadIdx.x;
  const float* zp = Z + (size_t)bi * L_ * E_ + e;
  float s = 0.f;
  for (int l = 0; l < L_; ++l) s += zp[(size_t)l * E_];
  DESC[bi * E_ + e] = s * (1.f / L_);
}

// ---- K3: g = d@gam_W+gam_b, bb = d@bet_W+bet_b ----
__global__ void k_gmod(const float* __restrict__ DESC, const float* __restrict__ gam_W,
                       const float* __restrict__ gam_b, const float* __restrict__ bet_W,
                       const float* __restrict__ bet_b, float* __restrict__ GMOD,
                       float* __restrict__ BMOD) {
  int i = blockIdx.x, b = blockIdx.y;
  int bi = b * NM_ + i;
  int e = threadIdx.x;
  __shared__ float d[E_];
  d[e] = DESC[bi * E_ + e];
  __syncthreads();
  float g = gam_b[i * E_ + e], bb = bet_b[i * E_ + e];
  for (int f = 0; f < E_; ++f) {
    g += d[f] * gam_W[((size_t)i * E_ + f) * E_ + e];
    bb += d[f] * bet_W[((size_t)i * E_ + f) * E_ + e];
  }
  GMOD[bi * E_ + e] = g;
  BMOD[bi * E_ + e] = bb;
}

// ---- K4: per-(b,i) instance-norm stats of x over 32768 voxels ----
__global__ void k_xstat(const float* __restrict__ x0, const float* __restrict__ x1,
                        const float* __restrict__ x2, const float* __restrict__ x3,
                        float* __restrict__ XSTAT) {
  int i = blockIdx.x, b = blockIdx.y;
  int bi = b * NM_ + i;
  const float* xs[4] = {x0, x1, x2, x3};
  const float* xp = xs[i] + (size_t)b * S_;
  __shared__ float sh[256], sh2[256];
  int t = threadIdx.x;
  float s = 0.f, q = 0.f;
  for (int k = t; k < S_; k += 256) {
    float v = xp[k];
    s += v;
    q += v * v;
  }
  sh[t] = s;
  sh2[t] = q;
  __syncthreads();
  for (int o = 128; o >= 1; o >>= 1) {
    if (t < o) { sh[t] += sh[t + o]; sh2[t] += sh2[t + o]; }
    __syncthreads();
  }
  if (t == 0) {
    float m = sh[0] / S_;
    float v = sh2[0] / S_ - m * m;
    XSTAT[bi * 2] = m;
    XSTAT[bi * 2 + 1] = rsqrtf(v + 1e-5f);
  }
}

// ---- K5: collapsed bypass scalar field u1 = bc1(bn(inorm(x))) ----
__global__ void k_u1(const float* __restrict__ x0, const float* __restrict__ x1,
                     const float* __restrict__ x2, const float* __restrict__ x3,
                     const float* __restrict__ XSTAT, const float* __restrict__ bn_w,
                     const float* __restrict__ bn_b, const float* __restrict__ bc1_w,
                     const float* __restrict__ bc1_b, float* __restrict__ U1) {
  size_t idx = (size_t)blockIdx.x * blockDim.x + threadIdx.x;
  if (idx >= (size_t)B_ * NM_ * S_) return;
  int s = (int)(idx & (S_ - 1));
  int bi = (int)(idx >> 15);
  int b = bi >> 2, i = bi & 3;
  const float* xs[4] = {x0, x1, x2, x3};
  float xv = xs[i][(size_t)b * S_ + s];
  float m = XSTAT[bi * 2], r = XSTAT[bi * 2 + 1];
  float u = (xv - m) * r * bn_w[i] + bn_b[i];
  U1[idx] = u * bc1_w[i] + bc1_b[i];
}

// ---- K6: layernorm + gamma/beta modulation -> mamba input ----
__global__ void k_lnmod(const float* __restrict__ Z, const float* __restrict__ ln_w,
                        const float* __restrict__ ln_b, const float* __restrict__ GMOD,
                        const float* __restrict__ BMOD, float* __restrict__ SMOD) {
  int l = blockIdx.x, i = blockIdx.y, b = blockIdx.z;
  int bi = b * NM_ + i;
  int t = threadIdx.x;
  __shared__ float sh[E_];
  size_t row = ((size_t)bi * L_ + l) * E_;
  float v = Z[row + t];
  sh[t] = v;
  __syncthreads();
  for (int o = 32; o >= 1; o >>= 1) {
    if (t < o) sh[t] += sh[t + o];
    __syncthreads();
  }
  float m = sh[0] * (1.f / E_);
  __syncthreads();
  float dv = v - m;
  sh[t] = dv * dv;
  __syncthreads();
  for (int o = 32; o >= 1; o >>= 1) {
    if (t < o) sh[t] += sh[t + o];
    __syncthreads();
  }
  float var = sh[0] * (1.f / E_);
  float xh = dv * rsqrtf(var + 1e-5f) * ln_w[i * E_ + t] + ln_b[i * E_ + t];
  SMOD[row + t] = GMOD[bi * E_ + t] * xh + BMOD[bi * E_ + t];
}

// ---- generic batched WMMA GEMM: C[bi] = A[bi] @ Bw[bi % bMod] ----
__global__ void k_gemm(const float* __restrict__ Ab, int lda, unsigned long long aStr,
                       const float* __restrict__ Bb, int ldb, unsigned long long bStr,
                       int bMod, float* __restrict__ Cb, int ldc,
                       unsigned long long cStr, int K) {
  int tn = blockIdx.x, tm = blockIdx.y, bi = blockIdx.z;
  const float* A = Ab + (size_t)bi * aStr + (size_t)tm * 16 * lda;
  const float* Bm = Bb + (size_t)(bi % bMod) * bStr + (size_t)tn * 16;
  float* C = Cb + (size_t)bi * cStr + (size_t)tm * 16 * ldc + (size_t)tn * 16;
  v8f acc = {};
  acc = wmma_k_f32(A, lda, Bm, ldb, K, acc);
  wmma_store(C, ldc, acc);
}

// ---- K8: causal depthwise conv (DC=4) + SiLU ----
__global__ void k_conv(const float* __restrict__ XZb, const float* __restrict__ conv_w,
                       const float* __restrict__ conv_b, float* __restrict__ XC) {
  size_t idx = (size_t)blockIdx.x * 256 + threadIdx.x;
  if (idx >= (size_t)B_ * NM_ * L_ * DI_) return;
  int d = (int)(idx & (DI_ - 1));
  int l = (int)((idx >> 7) & (L_ - 1));
  int bi = (int)(idx >> 19);
  int i = bi & 3;
  float s = conv_b[i * DI_ + d];
#pragma unroll
  for (int k = 0; k < DC_; ++k) {
    int lk = l - (DC_ - 1) + k;
    if (lk >= 0) s += conv_w[(i * DC_ + k) * DI_ + d] * XZb[((size_t)bi * L_ + lk) * 256 + d];
  }
  XC[idx] = siluf(s);
}

// ---- K9: xdb = xc @ m_x_W (128x36); split dt/B/C; dt = softplus(dt@m_dt_W + b) ----
__global__ void k_xdbdt(const float* __restrict__ XC, const float* __restrict__ m_x_W,
                        const float* __restrict__ m_dt_W, const float* __restrict__ m_dt_b,
                        float* __restrict__ BC, float* __restrict__ DT) {
  int l = blockIdx.x, i = blockIdx.y, b = blockIdx.z;
  int bi = b * NM_ + i;
  int t = threadIdx.x;  // 128
  __shared__ float xcr[DI_];
  __shared__ float xdb[DTR_ + 2 * DS_];
  size_t row = (size_t)bi * L_ + l;
  xcr[t] = XC[row * DI_ + t];
  __syncthreads();
  if (t < DTR_ + 2 * DS_) {
    float s = 0.f;
    const float* w = m_x_W + (size_t)i * DI_ * (DTR_ + 2 * DS_) + t;
    for (int dd = 0; dd < DI_; ++dd) s += xcr[dd] * w[(size_t)dd * (DTR_ + 2 * DS_)];
    xdb[t] = s;
  }
  __syncthreads();
  {
    float s = m_dt_b[i * DI_ + t];
    const float* w = m_dt_W + (size_t)i * DTR_ * DI_ + t;
#pragma unroll
    for (int r = 0; r < DTR_; ++r) s += xdb[r] * w[(size_t)r * DI_];
    DT[row * DI_ + t] = softplusf(s);
  }
  if (t >= DTR_ && t < DTR_ + 2 * DS_) BC[row * 32 + (t - DTR_)] = xdb[t];
}

// ---- K10: selective scan; state in registers; 16-lane shuffle reduce ----
__global__ void k_scan(const float* __restrict__ DT, const float* __restrict__ XC,
                       const float* __restrict__ BC, const float* __restrict__ Alog,
                       float* __restrict__ YS) {
  int i = blockIdx.x, b = blockIdx.y;
  int bi = b * NM_ + i;
  int t = threadIdx.x;  // 1024
  int n = t & 15;
  int dg = t >> 4;       // 0..63
  int d0 = dg, d1 = dg + 64;
  float A0 = -__expf(Alog[((size_t)i * DI_ + d0) * DS_ + n]);
  float A1 = -__expf(Alog[((size_t)i * DI_ + d1) * DS_ + n]);
  const float* dtp = DT + (size_t)bi * L_ * DI_;
  const float* xcp = XC + (size_t)bi * L_ * DI_;
  const float* bcp = BC + (size_t)bi * L_ * 32;
  float* ysp = YS + (size_t)bi * L_ * DI_;
  float h0 = 0.f, h1 = 0.f;
  for (int l = 0; l < L_; ++l) {
    size_t ro = (size_t)l * DI_;
    float dt0 = dtp[ro + d0], dt1 = dtp[ro + d1];
    float xv0 = xcp[ro + d0], xv1 = xcp[ro + d1];
    float Bv = bcp[l * 32 + n];
    float Cv = bcp[l * 32 + 16 + n];
    h0 = __expf(dt0 * A0) * h0 + dt0 * Bv * xv0;
    h1 = __expf(dt1 * A1) * h1 + dt1 * Bv * xv1;
    float p0 = h0 * Cv, p1 = h1 * Cv;
#pragma unroll
    for (int o = 8; o >= 1; o >>= 1) {
      p0 += __shfl_xor(p0, o, 16);
      p1 += __shfl_xor(p1, o, 16);
    }
    if (n == 0) {
      ysp[ro + d0] = p0;
      ysp[ro + d1] = p1;
    }
  }
}

// ---- K11: y = (ys + D*xc) * silu(zg), in place on YS ----
__global__ void k_gate(const float* __restrict__ XZb, const float* __restrict__ XC,
                       const float* __restrict__ m_D, float* __restrict__ YS) {
  size_t idx = (size_t)blockIdx.x * 256 + threadIdx.x;
  if (idx >= (size_t)B_ * NM_ * L_ * DI_) return;
  int d = (int)(idx & (DI_ - 1));
  int l = (int)((idx >> 7) & (L_ - 1));
  int bi = (int)(idx >> 19);
  int i = bi & 3;
  float zg = XZb[((size_t)bi * L_ + l) * 256 + DI_ + d];
  YS[idx] = (YS[idx] + m_D[i * DI_ + d] * XC[idx]) * siluf(zg);
}

// ---- K13: softmax(DD^T), U/V low-rank, Ceff[b,i] = sum_j A[b,i,j] * U_j V_j^T ----
__global__ void k_cross(const float* __restrict__ DESC, const float* __restrict__ U_W,
                        const float* __restrict__ U_b, const float* __restrict__ V_W,
                        const float* __restrict__ V_b, float* __restrict__ CEFF) {
  int b = blockIdx.x;
  int t = threadIdx.x;  // 256
  __shared__ float d[NM_][E_];
  __shared__ float Am[NM_][NM_];
  __shared__ float Us[E_ * R_];
  __shared__ float Vs[E_ * R_];
  d[t >> 6][t & 63] = DESC[(b * NM_ + (t >> 6)) * E_ + (t & 63)];
  __syncthreads();
  if (t < 16) {
    int ii = t >> 2, jj = t & 3;
    float s = 0.f;
    for (int e = 0; e < E_; ++e) s += d[ii][e] * d[jj][e];
    Am[ii][jj] = s;
  }
  __syncthreads();
  if (t < 4) {
    float mx = Am[t][0];
    for (int j = 1; j < 4; ++j) mx = fmaxf(mx, Am[t][j]);
    float ex[4], den = 0.f;
    for (int j = 0; j < 4; ++j) { ex[j] = __expf(Am[t][j] - mx); den += ex[j]; }
    for (int j = 0; j < 4; ++j) Am[t][j] = ex[j] / den;
  }
  __syncthreads();
  float acc[NM_][16];
#pragma unroll
  for (int ii = 0; ii < NM_; ++ii)
#pragma unroll
    for (int q = 0; q < 16; ++q) acc[ii][q] = 0.f;
  for (int j = 0; j < NM_; ++j) {
    for (int er = t; er < E_ * R_; er += 256) {
      float su = U_b[j * E_ * R_ + er], sv = V_b[j * E_ * R_ + er];
      const float* uw = U_W + (size_t)j * E_ * E_ * R_ + er;
      const float* vw = V_W + (size_t)j * E_ * E_ * R_ + er;
      for (int f = 0; f < E_; ++f) {
        su += d[j][f] * uw[(size_t)f * E_ * R_];
        sv += d[j][f] * vw[(size_t)f * E_ * R_];
      }
      Us[er] = su;
      Vs[er] = sv;
    }
    __syncthreads();
    for (int q = 0; q < 16; ++q) {
      int p = t * 16 + q;
      int e = p >> 6, f = p & 63;
      float s = 0.f;
      for (int r = 0; r < R_; ++r) s += Us[e * R_ + r] * Vs[f * R_ + r];
#pragma unroll
      for (int ii = 0; ii < NM_; ++ii) acc[ii][q] += Am[ii][j] * s;
    }
    __syncthreads();
  }
  for (int ii = 0; ii < NM_; ++ii)
    for (int q = 0; q < 16; ++q) {
      int p = t * 16 + q;
      int e = p >> 6, f = p & 63;
      CEFF[((size_t)(b * NM_ + ii) * E_ + e) * E_ + f] = acc[ii][q];
    }
}

// ---- K14: yfin = z @ D_W + D_b + Hs @ Ceff (dual WMMA GEMM) ----
__global__ void k_yfin(const float* __restrict__ Z, const float* __restrict__ HS,
                       const float* __restrict__ D_W, const float* __restrict__ D_b,
                       const float* __restrict__ CEFF, float* __restrict__ YF) {
  int tn = blockIdx.x, tm = blockIdx.y, bi = blockIdx.z;
  int i = bi & 3;
  v8f acc = {};
  acc = wmma_k_f32(Z + (size_t)bi * L_ * E_ + (size_t)tm * 16 * E_, E_,
                   D_W + (size_t)i * E_ * E_ + tn * 16, E_, E_, acc);
  acc = wmma_k_f32(HS + (size_t)bi * L_ * E_ + (size_t)tm * 16 * E_, E_,
                   CEFF + (size_t)bi * E_ * E_ + tn * 16, E_, E_, acc);
  int lane = threadIdx.x & 31;
  int n = lane & 15, mh = lane >> 4;
  float bv = D_b[i * E_ + tn * 16 + n];
  float* C = YF + (size_t)bi * L_ * E_ + (size_t)tm * 16 * E_ + tn * 16;
#pragma unroll
  for (int r = 0; r < 8; ++r) C[(size_t)(mh * 8 + r) * E_ + n] = acc[r] + bv;
}

// ---- K15: output inorm stats per (b,i,e): mean and rstd*on_w ----
__global__ void k_stats(const float* __restrict__ YF, const float* __restrict__ on_w,
                        float* __restrict__ STATS) {
  int e = blockIdx.x, i = blockIdx.y, b = blockIdx.z;
  int bi = b * NM_ + i;
  int t = threadIdx.x;
  __shared__ float sh[256], sh2[256];
  const float* yp = YF + (size_t)bi * L_ * E_ + e;
  float s = 0.f, q = 0.f;
  for (int l = t; l < L_; l += 256) {
    float v = yp[(size_t)l * E_];
    s += v;
    q += v * v;
  }
  sh[t] = s;
  sh2[t] = q;
  __syncthreads();
  for (int o = 128; o >= 1; o >>= 1) {
    if (t < o) { sh[t] += sh[t + o]; sh2[t] += sh2[t + o]; }
    __syncthreads();
  }
  if (t == 0) {
    float m = sh[0] / L_;
    float v = sh2[0] / L_ - m * m;
    STATS[(bi * E_ + e) * 2] = m;
    STATS[(bi * E_ + e) * 2 + 1] = rsqrtf(v + 1e-5f) * on_w[e];
  }
}

// ---- K16: fused norm + silu(byp) gate + 1x1 conv + sum over modules (WMMA) ----
__global__ void k_fuse(const float* __restrict__ YF, const float* __restrict__ STATS,
                       const float* __restrict__ on_b, const float* __restrict__ U1,
                       const float* __restrict__ bc2_w, const float* __restrict__ bc2_b,
                       const float* __restrict__ pp_w, const float* __restrict__ pp_b,
                       float* __restrict__ out) {
  int tn = blockIdx.x, tm = blockIdx.y, b = blockIdx.z;
  int lane = threadIdx.x & 31;
  int mr = lane & 15;
  int kh = lane >> 4;
  int srow = tm * 16 + mr;  // this lane's spatial row for A
  int zc = srow >> 10, yc = (srow >> 5) & 31, xc = srow & 31;
  int lpat = (zc >> 1) * 256 + (yc >> 1) * 16 + (xc >> 1);
  v8f acc = {};
  for (int i = 0; i < NM_; ++i) {
    int bi = b * NM_ + i;
    float u1v = U1[(size_t)bi * S_ + srow];
    const float* yrow = YF + ((size_t)bi * L_ + lpat) * E_;
    const float* st = STATS + (size_t)bi * E_ * 2;
    for (int k0 = 0; k0 < E_; k0 += 4) {
      int ka = k0 + 2 * kh;
      v2f a, bv;
#pragma unroll
      for (int q = 0; q < 2; ++q) {
        int c = ka + q;
        float byp = siluf(u1v * bc2_w[i * E_ + c] + bc2_b[i * E_ + c]);
        float vh = (yrow[c] - st[c * 2]) * st[c * 2 + 1] + on_b[c];
        float av = byp * vh;
        float bw = pp_w[((size_t)i * E_ + (tn * 16 + mr)) * E_ + c];
        if (q == 0) { a.x = av; bv.x = bw; } else { a.y = av; bv.y = bw; }
      }
      acc = wmma4(a, bv, acc);
    }
  }
  int o = tn * 16 + (lane & 15);
  float bias = 0.f;
#pragma unroll
  for (int i = 0; i < NM_; ++i) bias += pp_b[i * E_ + o];
  int mh = lane >> 4;
#pragma unroll
  for (int r = 0; r < 8; ++r) {
    int sp = tm * 16 + mh * 8 + r;
    out[((size_t)b * E_ + o) * S_ + sp] = acc[r] + bias;
  }
}

extern "C" void kernel_launch(void* const* d_in, const int* in_sizes, int n_in,
                              void* d_out, int out_size, void* d_ws, size_t ws_size,
                              hipStream_t stream) {
  (void)in_sizes; (void)n_in; (void)out_size; (void)ws_size;
  const float* x0 = (const float*)d_in[0];
  const float* x1 = (const float*)d_in[1];
  const float* x2 = (const float*)d_in[2];
  const float* x3 = (const float*)d_in[3];
  const float* in_W   = (const float*)d_in[4];
  const float* in_b   = (const float*)d_in[5];
  const float* ln_w   = (const float*)d_in[6];
  const float* ln_b   = (const float*)d_in[7];
  const float* gam_W  = (const float*)d_in[8];
  const float* gam_b  = (const float*)d_in[9];
  const float* bet_W  = (const float*)d_in[10];
  const float* bet_b  = (const float*)d_in[11];
  const float* m_in_W = (const float*)d_in[12];
  const float* m_conv_w = (const float*)d_in[13];
  const float* m_conv_b = (const float*)d_in[14];
  const float* m_x_W  = (const float*)d_in[15];
  const float* m_dt_W = (const float*)d_in[16];
  const float* m_dt_b = (const float*)d_in[17];
  const float* m_Alog = (const float*)d_in[18];
  const float* m_D    = (const float*)d_in[19];
  const float* m_out_W = (const float*)d_in[20];
  const float* U_W    = (const float*)d_in[21];
  const float* U_b    = (const float*)d_in[22];
  const float* V_W    = (const float*)d_in[23];
  const float* V_b    = (const float*)d_in[24];
  const float* D_W    = (const float*)d_in[25];
  const float* D_b    = (const float*)d_in[26];
  const float* on_w   = (const float*)d_in[27];
  const float* on_b   = (const float*)d_in[28];
  const float* bn_w   = (const float*)d_in[29];
  const float* bn_b   = (const float*)d_in[30];
  const float* bc1_w  = (const float*)d_in[31];
  const float* bc1_b  = (const float*)d_in[32];
  const float* bc2_w  = (const float*)d_in[33];
  const float* bc2_b  = (const float*)d_in[34];
  const float* pp_w   = (const float*)d_in[35];
  const float* pp_b   = (const float*)d_in[36];

  // ---- workspace layout (floats) ----
  float* W0 = (float*)d_ws;
  const size_t NZ = (size_t)B_ * NM_ * L_ * E_;         // 2,097,152
  float* Z    = W0;
  float* SMOD = Z + NZ;
  float* XZb  = SMOD + NZ;                              // B*NM*L*256
  float* XC   = XZb + (size_t)B_ * NM_ * L_ * 256;      // B*NM*L*128
  float* DTb  = XC + (size_t)B_ * NM_ * L_ * DI_;
  float* BCb  = DTb + (size_t)B_ * NM_ * L_ * DI_;      // B*NM*L*32
  float* YSb  = BCb + (size_t)B_ * NM_ * L_ * 32;       // B*NM*L*128
  float* HSb  = YSb + (size_t)B_ * NM_ * L_ * DI_;      // NZ
  float* YFb  = HSb + NZ;                               // NZ
  float* U1   = YFb + NZ;                               // B*NM*S
  float* DESC = U1 + (size_t)B_ * NM_ * S_;
  float* GMOD = DESC + (size_t)B_ * NM_ * E_;
  float* BMOD = GMOD + (size_t)B_ * NM_ * E_;
  float* CEFF = BMOD + (size_t)B_ * NM_ * E_;           // B*NM*E*E
  float* STATS = CEFF + (size_t)B_ * NM_ * E_ * E_;     // B*NM*E*2
  float* XSTAT = STATS + (size_t)B_ * NM_ * E_ * 2;     // B*NM*2

  k_patch<<<dim3(L_, NM_, B_), 64, 0, stream>>>(x0, x1, x2, x3, in_W, in_b, Z);
  k_desc<<<dim3(NM_, B_), 64, 0, stream>>>(Z, DESC);
  k_gmod<<<dim3(NM_, B_), 64, 0, stream>>>(DESC, gam_W, gam_b, bet_W, bet_b, GMOD, BMOD);
  k_xstat<<<dim3(NM_, B_), 256, 0, stream>>>(x0, x1, x2, x3, XSTAT);
  {
    size_t n = (size_t)B_ * NM_ * S_;
    k_u1<<<(unsigned)((n + 255) / 256), 256, 0, stream>>>(x0, x1, x2, x3, XSTAT, bn_w,
                                                          bn_b, bc1_w, bc1_b, U1);
  }
  k_lnmod<<<dim3(L_, NM_, B_), 64, 0, stream>>>(Z, ln_w, ln_b, GMOD, BMOD, SMOD);
  // xz = smod @ m_in_W  (M=4096, K=64, N=256)
  k_gemm<<<dim3(16, L_ / 16, B_ * NM_), 32, 0, stream>>>(
      SMOD, E_, (unsigned long long)L_ * E_,
      m_in_W, 256, (unsigned long long)E_ * 256, NM_,
      XZb, 256, (unsigned long long)L_ * 256, E_);
  {
    size_t n = (size_t)B_ * NM_ * L_ * DI_;
    k_conv<<<(unsigned)((n + 255) / 256), 256, 0, stream>>>(XZb, m_conv_w, m_conv_b, XC);
  }
  k_xdbdt<<<dim3(L_, NM_, B_), 128, 0, stream>>>(XC, m_x_W, m_dt_W, m_dt_b, BCb, DTb);
  k_scan<<<dim3(NM_, B_), 1024, 0, stream>>>(DTb, XC, BCb, m_Alog, YSb);
  {
    size_t n = (size_t)B_ * NM_ * L_ * DI_;
    k_gate<<<(unsigned)((n + 255) / 256), 256, 0, stream>>>(XZb, XC, m_D, YSb);
  }
  // Hs = y @ m_out_W  (M=4096, K=128, N=64)
  k_gemm<<<dim3(4, L_ / 16, B_ * NM_), 32, 0, stream>>>(
      YSb, DI_, (unsigned long long)L_ * DI_,
      m_out_W, E_, (unsigned long long)DI_ * E_, NM_,
      HSb, E_, (unsigned long long)L_ * E_, DI_);
  k_cross<<<B_, 256, 0, stream>>>(DESC, U_W, U_b, V_W, V_b, CEFF);
  k_yfin<<<dim3(4, L_ / 16, B_ * NM_), 32, 0, stream>>>(Z, HSb, D_W, D_b, CEFF, YFb);
  k_stats<<<dim3(E_, NM_, B_), 256, 0, stream>>>(YFb, on_w, STATS);
  k_fuse<<<dim3(4, S_ / 16, B_), 32, 0, stream>>>(YFb, STATS, on_b, U1, bc2_w, bc2_b,
                                                  pp_w, pp_b, (float*)d_out);
}